// JointMembership_FuzzyLayer_JAFFE_NoPool_89816356094179
// MI455X (gfx1250) — compile-verified
//
#include <hip/hip_runtime.h>

typedef float v2f __attribute__((ext_vector_type(2)));
typedef float v8f __attribute__((ext_vector_type(8)));

// Problem constants from the reference: x is (B,1,32,32) -> 1024 features,
// 512 pairs, 7 classes. Output (B, 1024, 14? no ->) (B, 512*2, 7).
// Row r = b*512 + p produces 14 contiguous outputs at out[14*r .. 14*r+14).
#define FEATS   1024
#define NPAIRS  512
#define NCLS    7

__global__ __launch_bounds__(256) void qcirc_wmma_kernel(
    const float* __restrict__ x,      // [B*1024]
    const int*   __restrict__ pairs,  // [512*2]
    const float* __restrict__ theta,  // [7]
    float*       __restrict__ out,    // [B*512*14]
    int n_rows)                        // B*512 (multiple of 16)
{
    const int lane  = threadIdx.x & 31;
    const int wave  = threadIdx.x >> 5;
    const int gwave = blockIdx.x * (blockDim.x >> 5) + wave;
    const int nwav  = gridDim.x * (blockDim.x >> 5);

    const int  n  = lane & 15;     // column within tile / row-in-tile for A
    const bool hi = lane >= 16;    // upper half-wave -> K=2,3 fragments

    // ---- Loop-invariant B fragment: W is 4x16 (K x N), N = q*7 + c ----
    // lanes 0-15 : VGPR0 = W[0][n], VGPR1 = W[1][n]
    // lanes 16-31: VGPR0 = W[2][n], VGPR1 = W[3][n]
    float th = theta[(n < NCLS) ? n : 0];
    float st, ct;
    __sincosf(th, &st, &ct);
    v2f bfrag;
    if (!hi) {
        bfrag.x = (n < 2 * NCLS) ? 0.5f : 0.0f;        // W[0][n] = 1/2
        bfrag.y = (n < NCLS) ? 0.5f * ct : 0.0f;       // W[1][n] = cos(th)/2
    } else {
        bfrag.x = (n < NCLS) ? -0.5f * st : 0.0f;      // W[2][n] = -sin(th)/2
        bfrag.y = (n >= NCLS && n < 2 * NCLS) ? 0.5f : 0.0f; // W[3][n] = 1/2
    }

    const int ntiles = n_rows >> 4;
    for (int t = gwave; t < ntiles; t += nwav) {       // wave-uniform loop
        const int row = (t << 4) + n;                  // this lane's matrix row
        const int b   = row >> 9;                      // / NPAIRS (512)
        const int p   = row & (NPAIRS - 1);
        const int i0  = pairs[2 * p];
        const int i1  = pairs[2 * p + 1];
        const float av = x[b * FEATS + i0];
        const float bv = x[b * FEATS + i1];

        float sa, ca, sb, cb;
        __sincosf(av, &sa, &ca);
        __sincosf(bv, &sb, &cb);

        // A fragment (16x4 f32): lanes 0-15 carry K=0,1; lanes 16-31 K=2,3
        v2f afrag;
        if (!hi) { afrag.x = 1.0f;    afrag.y = ca;      }
        else     { afrag.x = sa * sb; afrag.y = ca * cb; }

        v8f acc = {};
        // D(16x16,f32) = A(16x4) * W(4x16) + 0   -> v_wmma_f32_16x16x4_f32
        acc = __builtin_amdgcn_wmma_f32_16x16x4_f32(
            /*neg_a=*/false, afrag, /*neg_b=*/false, bfrag,
            /*c_mod=*/(short)0, acc, /*reuse_a=*/false, /*reuse_b=*/false);

        // D layout: VGPR v of lane -> element (M = v + 8*hi, N = n).
        // out flat index = 14*(tile_row0 + M) + n ; N=14,15 are padding.
        if (n < 2 * NCLS) {
            float* o = out + (long)(2 * NCLS) * ((long)(t << 4) + (hi ? 8 : 0)) + n;
            #pragma unroll
            for (int v = 0; v < 8; ++v) {
                __builtin_nontemporal_store(acc[v], o + (2 * NCLS) * v);
            }
        }
    }
}

extern "C" void kernel_launch(void* const* d_in, const int* in_sizes, int n_in,
                              void* d_out, int out_size, void* d_ws, size_t ws_size,
                              hipStream_t stream) {
    const float* x     = (const float*)d_in[0];   // (B,1,32,32) f32
    const int*   pairs = (const int*)  d_in[1];   // (512,2) i32
    const float* theta = (const float*)d_in[2];   // (7,) f32
    float*       out   = (float*)      d_out;     // (B, 1024, 7) f32

    const int B      = in_sizes[0] / FEATS;       // 8192
    const int n_rows = B * NPAIRS;                // 4,194,304 rows, 262,144 tiles

    // 1024 blocks x 8 waves = 8192 waves; each wave streams ~32 tiles.
    dim3 block(256);
    dim3 grid(1024);
    qcirc_wmma_kernel<<<grid, block, 0, stream>>>(x, pairs, theta, out, n_rows);
}